// TopKRouter_87402584474273
// MI455X (gfx1250) — compile-verified
//
#include <hip/hip_runtime.h>
#include <hip/hip_bf16.h>

typedef float v2f __attribute__((ext_vector_type(2)));
typedef float v8f __attribute__((ext_vector_type(8)));

constexpr int   T    = 16384;   // tokens
constexpr int   D    = 2048;    // input dim
constexpr int   E    = 64;      // experts
constexpr float COEF = 64.0f * 0.1f / (16384.0f * 16384.0f * 2.0f);

constexpr int WAVES        = 8;                     // waves per block
constexpr int TOK_PER_WAVE = 16;                    // one 16-row WMMA M-tile
constexpr int TOK_PER_BLK  = WAVES * TOK_PER_WAVE;  // 128 tokens per block
constexpr int LDS_STRIDE   = E + 1;                 // 65 -> conflict-free column reads

// ---------------------------------------------------------------------------
// zero the workspace (agg[64], cnt[64])
// ---------------------------------------------------------------------------
__global__ void TopKRouter_zero_kernel(float* ws) {
    ws[threadIdx.x] = 0.0f;
}

// ---------------------------------------------------------------------------
// gating GEMM (fp32 WMMA) + softmax + top-2 + block-level reductions
// ---------------------------------------------------------------------------
__global__ __launch_bounds__(256) void TopKRouter_gemm_kernel(
    const float* __restrict__ x,      // [T, D]
    const float* __restrict__ w,      // [E, D]
    float* __restrict__ out,          // [2T probs | 2T indices | 1 aux]
    float* __restrict__ g_agg,        // [E]
    float* __restrict__ g_cnt)        // [E]
{
    __shared__ float lds[TOK_PER_BLK * LDS_STRIDE];   // logits, then probs
    __shared__ float cnt_lds[E];

    const int tid   = threadIdx.x;
    const int wave  = tid >> 5;
    const int lane  = tid & 31;
    const int lhalf = lane >> 4;      // 0: lanes 0-15, 1: lanes 16-31
    const int lmod  = lane & 15;
    const int m0    = blockIdx.x * TOK_PER_BLK + wave * TOK_PER_WAVE;

    if (tid < E) cnt_lds[tid] = 0.0f;

    // A 16x4 fp32 fragment layout: lane(l%16)=M row, VGPR0/1 = K = 2*(l/16)+{0,1}
    // B 4x16  fp32 fragment layout: lane(l%16)=N col, VGPR0/1 = K = 2*(l/16)+{0,1}
    // B[k][n] = w[n][k]  (logits = x @ w^T), so per-lane B loads are contiguous k pairs.
    const float* arow  = x + (size_t)(m0 + lmod) * D + 2 * lhalf;
    const float* brow0 = w + (size_t)( 0 + lmod) * D + 2 * lhalf;
    const float* brow1 = w + (size_t)(16 + lmod) * D + 2 * lhalf;
    const float* brow2 = w + (size_t)(32 + lmod) * D + 2 * lhalf;
    const float* brow3 = w + (size_t)(48 + lmod) * D + 2 * lhalf;

    v8f acc0 = {}, acc1 = {}, acc2 = {}, acc3 = {};

#pragma unroll 4
    for (int k = 0; k < D; k += 4) {
        v2f a  = *(const v2f*)(arow  + k);
        v2f b0 = *(const v2f*)(brow0 + k);
        v2f b1 = *(const v2f*)(brow1 + k);
        v2f b2 = *(const v2f*)(brow2 + k);
        v2f b3 = *(const v2f*)(brow3 + k);
        acc0 = __builtin_amdgcn_wmma_f32_16x16x4_f32(false, a, false, b0, (short)0, acc0, false, false);
        acc1 = __builtin_amdgcn_wmma_f32_16x16x4_f32(false, a, false, b1, (short)0, acc1, false, false);
        acc2 = __builtin_amdgcn_wmma_f32_16x16x4_f32(false, a, false, b2, (short)0, acc2, false, false);
        acc3 = __builtin_amdgcn_wmma_f32_16x16x4_f32(false, a, false, b3, (short)0, acc3, false, false);
    }

    // C/D layout: VGPR j, lanes 0-15 -> M=j, lanes 16-31 -> M=j+8; N = lane%16
    const int mloc = wave * TOK_PER_WAVE + 8 * lhalf;
#pragma unroll
    for (int j = 0; j < 8; ++j) {
        float* dst = &lds[(mloc + j) * LDS_STRIDE + lmod];
        dst[0]  = acc0[j];
        dst[16] = acc1[j];
        dst[32] = acc2[j];
        dst[48] = acc3[j];
    }
    __syncthreads();

    // --- per-token softmax + top-2 (threads 0..127, one token each) ---
    if (tid < TOK_PER_BLK) {
        float* row = &lds[tid * LDS_STRIDE];

        float mx = row[0];
#pragma unroll
        for (int e = 1; e < E; ++e) mx = fmaxf(mx, row[e]);

        float sum = 0.0f;
        float b1v = -3.402823466e+38f, b2v = -3.402823466e+38f;
        int   b1i = 0, b2i = 0;
#pragma unroll
        for (int e = 0; e < E; ++e) {
            float v = row[e];
            sum += __expf(v - mx);
            if (v > b1v)      { b2v = b1v; b2i = b1i; b1v = v; b1i = e; }
            else if (v > b2v) { b2v = v; b2i = e; }
        }
        const float inv   = 1.0f / sum;
        const int   token = blockIdx.x * TOK_PER_BLK + tid;

        out[2 * token + 0]         = __expf(b1v - mx) * inv;
        out[2 * token + 1]         = __expf(b2v - mx) * inv;
        out[2 * T + 2 * token + 0] = (float)b1i;
        out[2 * T + 2 * token + 1] = (float)b2i;

        atomicAdd(&cnt_lds[b1i], 1.0f);
        atomicAdd(&cnt_lds[b2i], 1.0f);

        // overwrite own row with probabilities for the column reduction
#pragma unroll
        for (int e = 0; e < E; ++e) row[e] = __expf(row[e] - mx) * inv;
    }
    __syncthreads();

    // --- per-expert reductions: agg probs + top-2 counts (threads 0..63) ---
    if (tid < E) {
        float s = 0.0f;
        for (int t = 0; t < TOK_PER_BLK; ++t) s += lds[t * LDS_STRIDE + tid];
        atomicAdd(&g_agg[tid], s);
        atomicAdd(&g_cnt[tid], cnt_lds[tid]);
    }
}

// ---------------------------------------------------------------------------
// aux loss: sum(agg[e] * cnt[e]) * COEF, single wave32
// ---------------------------------------------------------------------------
__global__ void TopKRouter_finalize_kernel(const float* __restrict__ g_agg,
                                           const float* __restrict__ g_cnt,
                                           float* __restrict__ out) {
    const int lane = threadIdx.x;  // 0..31
    float v = g_agg[lane] * g_cnt[lane] + g_agg[lane + 32] * g_cnt[lane + 32];
#pragma unroll
    for (int off = 16; off > 0; off >>= 1) v += __shfl_down(v, off, 32);
    if (lane == 0) out[4 * T] = v * COEF;
}

// ---------------------------------------------------------------------------
extern "C" void kernel_launch(void* const* d_in, const int* in_sizes, int n_in,
                              void* d_out, int out_size, void* d_ws, size_t ws_size,
                              hipStream_t stream) {
    const float* x = (const float*)d_in[0];   // [16384, 2048] fp32
    const float* w = (const float*)d_in[1];   // [64, 2048]    fp32
    float* out   = (float*)d_out;             // 2T + 2T + 1 floats
    float* g_agg = (float*)d_ws;              // [64]
    float* g_cnt = g_agg + E;                 // [64]

    TopKRouter_zero_kernel<<<1, 2 * E, 0, stream>>>(g_agg);
    TopKRouter_gemm_kernel<<<T / TOK_PER_BLK, 256, 0, stream>>>(x, w, out, g_agg, g_cnt);
    TopKRouter_finalize_kernel<<<1, 32, 0, stream>>>(g_agg, g_cnt, out);
}